// ExplicitLiePE_24515673325835
// MI455X (gfx1250) — compile-verified
//
#include <hip/hip_runtime.h>
#include <math.h>

typedef __attribute__((ext_vector_type(2))) float v2f;
typedef __attribute__((ext_vector_type(8))) float v8f;

#define DH        64
#define LDA       66          // padded LDS row stride (floats) to avoid bank conflicts
#define NTHREADS  128         // 4 wave32s
#define TAYLOR_J  10

// 64x64 fp32 matmul D = A*B using v_wmma_f32_16x16x4_f32.
// 4 waves: wave w owns the 16-row band M=[16w,16w+16); each wave does 4 N-tiles.
// Accumulates in registers, internal barrier, then writes Tout (may alias A/B)
// scaled by `sc`, optionally accumulating into Eadd. Ends with a barrier.
__device__ __forceinline__ void mm64_scale_acc(const float* As, const float* Bs,
                                               float* Tout, float sc, float* Eadd,
                                               int lane, int wave) {
  const int l16   = lane & 15;
  const int khalf = (lane >> 4) << 1;          // lanes 16..31 handle K+2,K+3
  const int mrow  = (wave << 4) + l16;         // A row fed by this lane
  v8f acc[4];
#pragma unroll
  for (int n = 0; n < 4; ++n) {
    const int ncol = (n << 4) + l16;           // B column fed by this lane
    v8f c = {0.f, 0.f, 0.f, 0.f, 0.f, 0.f, 0.f, 0.f};
    for (int k0 = 0; k0 < DH; k0 += 4) {
      // A fragment: 16x4 tile, lane holds A[mrow][k0+khalf .. +1] (8B aligned)
      v2f a = *(const v2f*)(As + mrow * LDA + k0 + khalf);
      // B fragment: 4x16 tile, lane holds B[k0+khalf..+1][ncol]
      v2f b;
      b.x = Bs[(k0 + khalf) * LDA + ncol];
      b.y = Bs[(k0 + khalf + 1) * LDA + ncol];
      c = __builtin_amdgcn_wmma_f32_16x16x4_f32(false, a, false, b,
                                                (short)0, c, false, false);
    }
    acc[n] = c;
  }
  __syncthreads();                             // all reads of As/Bs complete
  const int crow0 = (wave << 4) + ((lane >> 4) << 3);  // C: lanes 16..31 -> M+8
#pragma unroll
  for (int n = 0; n < 4; ++n) {
    const int ncol = (n << 4) + l16;
#pragma unroll
    for (int v = 0; v < 8; ++v) {
      float t = acc[n][v] * sc;
      Tout[(crow0 + v) * LDA + ncol] = t;
      if (Eadd) Eadd[(crow0 + v) * LDA + ncol] += t;
    }
  }
  __syncthreads();
}

__global__ __launch_bounds__(NTHREADS)
void lie_pe_expm_kernel(const float* __restrict__ x,
                        const float* __restrict__ r_grid,
                        const float* __restrict__ L,
                        const float* __restrict__ P,
                        float* __restrict__ out) {
  __shared__ float M[DH * LDA];   // scaled generator
  __shared__ float T[DH * LDA];   // current Taylor term
  __shared__ float E[DH * LDA];   // exponential accumulator
  __shared__ float xv[DH];
  __shared__ float zv[DH];        // z = P_sp @ x
  __shared__ float red[DH];

  const int tid  = threadIdx.x;
  const int lane = tid & 31;
  const int wave = tid >> 5;
  const int bs   = blockIdx.x;

  const float r0 = r_grid[bs * 3 + 0];
  const float r1 = r_grid[bs * 3 + 1];
  const float r2 = r_grid[bs * 3 + 2];

  if (tid < DH) xv[tid] = x[bs * DH + tid];

  const float* L0 = L;
  const float* L1 = L + DH * DH;
  const float* L2 = L + 2 * DH * DH;

  // A[i][j] = sum_k r_k * 0.5*(L_k[i][j] - L_k[j][i])
  for (int idx = tid; idx < DH * DH; idx += NTHREADS) {
    const int i = idx >> 6, j = idx & 63;
    float a = 0.5f * (r0 * (L0[i * DH + j] - L0[j * DH + i]) +
                      r1 * (L1[i * DH + j] - L1[j * DH + i]) +
                      r2 * (L2[i * DH + j] - L2[j * DH + i]));
    M[i * LDA + j] = a;
  }
  __syncthreads();

  // z = P_sp @ x (identity in the reference setup; applied for generality)
  if (tid < DH) {
    float acc = 0.f;
    for (int j = 0; j < DH; ++j) acc += P[tid * DH + j] * xv[j];
    zv[tid] = acc;
  }

  // infinity norm of A for scaling-and-squaring
  if (tid < DH) {
    float s = 0.f;
    for (int j = 0; j < DH; ++j) s += fabsf(M[tid * LDA + j]);
    red[tid] = s;
  }
  __syncthreads();
  float nrm = 0.f;
  for (int i = 0; i < DH; ++i) nrm = fmaxf(nrm, red[i]);
  int sq = 0;
  if (nrm > 0.25f) sq = (int)ceilf(log2f(nrm * 4.0f));
  if (sq > 40) sq = 40;
  sq = __builtin_amdgcn_readfirstlane(sq);   // uniform -> scalar loop, EXEC stays full
  const float scale = exp2f((float)(-sq));

  // M *= 2^-sq ; T = M ; E = I + M
  for (int idx = tid; idx < DH * DH; idx += NTHREADS) {
    const int i = idx >> 6, j = idx & 63;
    float m = M[i * LDA + j] * scale;
    M[i * LDA + j] = m;
    T[i * LDA + j] = m;
    E[i * LDA + j] = m + (i == j ? 1.f : 0.f);
  }
  __syncthreads();

  // Taylor: T <- (T @ M)/j ; E += T   for j = 2..TAYLOR_J
  for (int jt = 2; jt <= TAYLOR_J; ++jt)
    mm64_scale_acc(T, M, T, 1.f / (float)jt, E, lane, wave);

  // Squaring: E <- E @ E, sq times (uniform scalar trip count)
  for (int it = 0; it < sq; ++it)
    mm64_scale_acc(E, E, E, 1.f, nullptr, lane, wave);

  // y = E @ z
  if (tid < DH) {
    float acc = 0.f;
    for (int j = 0; j < DH; ++j) acc += E[tid * LDA + j] * zv[j];
    out[bs * DH + tid] = acc;
  }
}

extern "C" void kernel_launch(void* const* d_in, const int* in_sizes, int n_in,
                              void* d_out, int out_size, void* d_ws, size_t ws_size,
                              hipStream_t stream) {
  const float* x = (const float*)d_in[0];   // [B,S,64]
  const float* r = (const float*)d_in[1];   // [B,S,3]
  const float* L = (const float*)d_in[2];   // [3,64,64]
  const float* P = (const float*)d_in[3];   // [64,64]
  float* out = (float*)d_out;               // [B,S,64]

  const int n_bs = in_sizes[0] / DH;        // B*S = 8192
  hipLaunchKernelGGL(lie_pe_expm_kernel, dim3(n_bs), dim3(NTHREADS), 0, stream,
                     x, r, L, P, out);
}